// TemporalTopographicalAttention_50130858279031
// MI455X (gfx1250) — compile-verified
//
#include <hip/hip_runtime.h>

// X [B=256, C=128, F=2048] f32, A [128, 2048] f32
#define B_DIM 256
#define C_DIM 128
#define F_DIM 2048
#define BDIM  256            // 8 wave32 waves

#define KCA   64             // phase A K-chunk width
#define LSTRA 68             // phase A LDS row stride (floats): 64 data + 4 pad
#define NCC   128            // phase C N-chunk width
#define LSTRC 132            // phase C / c_buf LDS row stride: 128 data + 4 pad

typedef float v2f __attribute__((ext_vector_type(2)));
typedef float v8f __attribute__((ext_vector_type(8)));
typedef unsigned int u32x4 __attribute__((ext_vector_type(4)));
typedef int i32x4 __attribute__((ext_vector_type(4)));
typedef int i32x8 __attribute__((ext_vector_type(8)));

// Tensor Data Mover: DMA a tile_h x tile_w f32 tile (row stride row_stride elems)
// from global into LDS at byte address lds_byte, inserting LDS padding every
// 2^(pad_int_code+1) dwords (writes the bank-conflict-padded layout directly).
__device__ __forceinline__ void tdm_load_2d(uint32_t lds_byte, const float* g,
                                            uint32_t tile_w, uint32_t tile_h,
                                            uint32_t row_stride,
                                            uint32_t pad_int_code, uint32_t pad_amt_code) {
  const uint64_t ga = (uint64_t)(uintptr_t)g;
  u32x4 g0;
  g0[0] = 1u;                                   // count=1 (valid user descriptor)
  g0[1] = lds_byte;                             // LDS destination byte address
  g0[2] = (uint32_t)ga;                         // global_addr[31:0]
  g0[3] = (uint32_t)(ga >> 32) | (2u << 30);    // global_addr[56:32] | type=2 (image)
  i32x8 g1;
  g1[0] = (int)((2u << 16)                      // data_size = 4 bytes
              | (1u << 20)                      // pad_enable
              | (pad_int_code << 22)            // pad interval (dword period)
              | (pad_amt_code << 25));          // pad amount (dwords-1)
  g1[1] = (int)((tile_w & 0xFFFFu) << 16);      // tensor_dim0[15:0]  (== tile_w)
  g1[2] = (int)((tile_w >> 16) | ((tile_h & 0xFFFFu) << 16)); // dim0 hi | tensor_dim1 lo
  g1[3] = (int)((tile_h >> 16) | (tile_w << 16));             // dim1 hi | tile_dim0
  g1[4] = (int)tile_h;                          // tile_dim1 | tile_dim2=0
  g1[5] = (int)row_stride;                      // tensor_dim0_stride[31:0]
  g1[6] = 0;                                    // stride hi | tensor_dim1_stride lo
  g1[7] = 0;
  const i32x4 gz4 = {0, 0, 0, 0};               // groups 2/3: unused (2-D tensor)
  const i32x8 gz8 = {0, 0, 0, 0, 0, 0, 0, 0};
  __builtin_amdgcn_tensor_load_to_lds(g0, g1, gz4, gz4, gz8, 0);
}

__global__ void __launch_bounds__(BDIM, 1)
sparse_attn_kernel(const float* __restrict__ X, const float* __restrict__ A,
                   float* __restrict__ Out, float* __restrict__ Wout) {
  extern __shared__ float lds[];
  float* c_buf  = lds;                          // 128 x LSTRC : logits -> sparsemax W
  float* region = lds + C_DIM * LSTRC;          // staging region (double buffered)
  // phase A layout inside region: x0 | x1 | a0 | a1, each 128 x LSTRA
  // phase C layout inside region: xc0 | xc1, each 128 x LSTRC
  const uint32_t lds_base   = (uint32_t)(uintptr_t)(void*)lds;   // LDS byte offset
  const uint32_t region_b   = lds_base + C_DIM * LSTRC * 4u;
  const uint32_t xoffA[2]   = {0u,               C_DIM * LSTRA * 4u};
  const uint32_t aoffA[2]   = {2u * C_DIM * LSTRA * 4u, 3u * C_DIM * LSTRA * 4u};
  const uint32_t xoffC[2]   = {0u,               C_DIM * LSTRC * 4u};

  const int tid  = threadIdx.x;
  const int lane = tid & 31;
  const int wave = tid >> 5;                    // tile-row strip owner (M = wave*16..+15)
  const int lo   = lane & 15;
  const int hi   = lane >> 4;

  const int b = blockIdx.x;
  const float* Xb = X + (size_t)b * C_DIM * F_DIM;

  // ============ Phase A: c = X[b] (128x2048) @ A^T -> 128x128 in LDS ============
  v8f acc[8];
#pragma unroll
  for (int nt = 0; nt < 8; ++nt) acc[nt] = (v8f){0.f,0.f,0.f,0.f,0.f,0.f,0.f,0.f};

  if (wave == 0) {                              // preload chunk 0 via TDM
    tdm_load_2d(region_b + xoffA[0], Xb, KCA, C_DIM, F_DIM, 5, 3);
    tdm_load_2d(region_b + aoffA[0], A,  KCA, C_DIM, F_DIM, 5, 3);
  }

  int cur = 0;
#pragma unroll 1
  for (int kc = 0; kc < F_DIM; kc += KCA) {
    if (wave == 0) {
      if (kc + KCA < F_DIM) {                   // prefetch next chunk into idle buffer
        tdm_load_2d(region_b + xoffA[cur ^ 1], Xb + kc + KCA, KCA, C_DIM, F_DIM, 5, 3);
        tdm_load_2d(region_b + aoffA[cur ^ 1], A  + kc + KCA, KCA, C_DIM, F_DIM, 5, 3);
        __builtin_amdgcn_s_wait_tensorcnt(2);   // current pair retired (in-order)
      } else {
        __builtin_amdgcn_s_wait_tensorcnt(0);
      }
    }
    __syncthreads();

    const float* xbuf = region + (xoffA[cur] >> 2);
    const float* abuf = region + (aoffA[cur] >> 2);
    const float* xrow = &xbuf[(wave * 16 + lo) * LSTRA + 2 * hi];
#pragma unroll 1
    for (int k0 = 0; k0 < KCA; k0 += 4) {
      v2f af;                                   // A-frag: M=lo, K=k0+2*hi+{0,1}
      af.x = xrow[k0];
      af.y = xrow[k0 + 1];
#pragma unroll
      for (int nt = 0; nt < 8; ++nt) {          // B-frag: B[k][n]=A[n][k] same striping
        const float* arow = &abuf[(nt * 16 + lo) * LSTRA + k0 + 2 * hi];
        v2f bf;
        bf.x = arow[0];
        bf.y = arow[1];
        acc[nt] = __builtin_amdgcn_wmma_f32_16x16x4_f32(
            false, af, false, bf, (short)0, acc[nt], false, false);
      }
    }
    __syncthreads();                            // all readers done before DMA reuses buf
    cur ^= 1;
  }

  // park logits: C/D layout -> M = r + 8*hi, N = lo
#pragma unroll
  for (int nt = 0; nt < 8; ++nt)
#pragma unroll
    for (int r = 0; r < 8; ++r)
      c_buf[(wave * 16 + 8 * hi + r) * LSTRC + nt * 16 + lo] = acc[nt][r];
  __syncthreads();

  // kick off phase-C chunk 0 DMA now; it overlaps the sparsemax phase below
  if (wave == 0)
    tdm_load_2d(region_b + xoffC[0], Xb, NCC, C_DIM, F_DIM, 6, 3);

  // ============ Phase B: row-wise sparsemax via tau bisection ============
  // tau solves sum(relu(z - tau)) == 1; after max-shift tau is in [-1, 0].
  float* Wb = Wout + (size_t)b * C_DIM * C_DIM;
#pragma unroll 1
  for (int rr = 0; rr < 16; ++rr) {
    const int row = wave * 16 + rr;
    float z0 = c_buf[row * LSTRC + lane];
    float z1 = c_buf[row * LSTRC + lane + 32];
    float z2 = c_buf[row * LSTRC + lane + 64];
    float z3 = c_buf[row * LSTRC + lane + 96];
    float m = fmaxf(fmaxf(z0, z1), fmaxf(z2, z3));
#pragma unroll
    for (int off = 16; off >= 1; off >>= 1)
      m = fmaxf(m, __shfl_xor(m, off, 32));
    z0 -= m; z1 -= m; z2 -= m; z3 -= m;

    float tlo = -1.0f, thi = 0.0f;
#pragma unroll 1
    for (int it = 0; it < 30; ++it) {
      const float tau = 0.5f * (tlo + thi);
      float s = fmaxf(z0 - tau, 0.f) + fmaxf(z1 - tau, 0.f) +
                fmaxf(z2 - tau, 0.f) + fmaxf(z3 - tau, 0.f);
#pragma unroll
      for (int off = 16; off >= 1; off >>= 1)
        s += __shfl_xor(s, off, 32);
      if (s >= 1.0f) tlo = tau; else thi = tau;    // s is wave-uniform
    }
    const float tau = 0.5f * (tlo + thi);
    const float w0 = fmaxf(z0 - tau, 0.f);
    const float w1 = fmaxf(z1 - tau, 0.f);
    const float w2 = fmaxf(z2 - tau, 0.f);
    const float w3 = fmaxf(z3 - tau, 0.f);
    c_buf[row * LSTRC + lane]      = w0;
    c_buf[row * LSTRC + lane + 32] = w1;
    c_buf[row * LSTRC + lane + 64] = w2;
    c_buf[row * LSTRC + lane + 96] = w3;
    Wb[(size_t)row * C_DIM + lane]      = w0;      // W output, coalesced
    Wb[(size_t)row * C_DIM + lane + 32] = w1;
    Wb[(size_t)row * C_DIM + lane + 64] = w2;
    Wb[(size_t)row * C_DIM + lane + 96] = w3;
  }
  __syncthreads();

  // ============ Phase C: out[b] = W (128x128) @ X[b] (128x2048) ============
  float* Ob = Out + (size_t)b * C_DIM * F_DIM;
  cur = 0;
#pragma unroll 1
  for (int fc = 0; fc < F_DIM; fc += NCC) {
    if (wave == 0) {
      if (fc + NCC < F_DIM) {
        tdm_load_2d(region_b + xoffC[cur ^ 1], Xb + fc + NCC, NCC, C_DIM, F_DIM, 6, 3);
        __builtin_amdgcn_s_wait_tensorcnt(1);    // current chunk retired
      } else {
        __builtin_amdgcn_s_wait_tensorcnt(0);
      }
    }
    __syncthreads();

    const float* xbuf = region + (xoffC[cur] >> 2);   // xbuf[k][n] = X[b][k][fc+n]
    const float* wrow = &c_buf[(wave * 16 + lo) * LSTRC + 2 * hi];
#pragma unroll 1
    for (int nt = 0; nt < 8; ++nt) {
      v8f o = (v8f){0.f,0.f,0.f,0.f,0.f,0.f,0.f,0.f};
#pragma unroll 1
      for (int k0 = 0; k0 < NCC; k0 += 4) {
        v2f af;                                  // A-frag from W
        af.x = wrow[k0];
        af.y = wrow[k0 + 1];
        v2f bf;                                  // B-frag: B[k][n] = xbuf[k][n]
        bf.x = xbuf[(k0 + 2 * hi) * LSTRC + nt * 16 + lo];
        bf.y = xbuf[(k0 + 2 * hi + 1) * LSTRC + nt * 16 + lo];
        o = __builtin_amdgcn_wmma_f32_16x16x4_f32(
            false, af, false, bf, (short)0, o, false, false);
      }
#pragma unroll
      for (int r = 0; r < 8; ++r)
        Ob[(size_t)(wave * 16 + 8 * hi + r) * F_DIM + fc + nt * 16 + lo] = o[r];
    }
    __syncthreads();
    cur ^= 1;
  }
}

extern "C" void kernel_launch(void* const* d_in, const int* in_sizes, int n_in,
                              void* d_out, int out_size, void* d_ws, size_t ws_size,
                              hipStream_t stream) {
  (void)in_sizes; (void)n_in; (void)out_size; (void)d_ws; (void)ws_size;
  const float* X = (const float*)d_in[0];   // [256,128,2048]
  const float* A = (const float*)d_in[1];   // [128,2048]
  float* Out  = (float*)d_out;                                   // [256,128,2048]
  float* Wout = Out + (size_t)B_DIM * C_DIM * F_DIM;             // [256,128,128]

  // c_buf (128*132) + staging region max(4*128*68, 2*128*132) floats  ~= 202 KB
  const size_t region_fl = 4u * C_DIM * LSTRA;           // 34816 floats (phase A worst)
  const size_t lds_bytes = ((size_t)C_DIM * LSTRC + region_fl) * sizeof(float);
  (void)hipFuncSetAttribute(reinterpret_cast<const void*>(sparse_attn_kernel),
                            hipFuncAttributeMaxDynamicSharedMemorySize, (int)lds_bytes);

  sparse_attn_kernel<<<B_DIM, BDIM, lds_bytes, stream>>>(X, A, Out, Wout);
}